// GraphSAGE_18322330484806
// MI455X (gfx1250) — compile-verified
//
#include <hip/hip_runtime.h>

// ---------------------------------------------------------------------------
// GraphSAGE (2x SAGEConv, aggregator='gcn') for MI455X / gfx1250 (wave32).
//
// Pipeline (all on `stream`):
//   deg,buf0 <- 0
//   scatter1: buf0[dst,:] += feat[src,:]; deg[dst] += 1          (f32 atomics)
//   gemm1:    buf1 = relu(((buf0+feat)/(deg+1)) @ W1 + b1)       (V_WMMA f32)
//   buf0 <- 0
//   scatter2: buf0[dst,:] += buf1[src,:]
//   gemm2:    out  = ((buf0+buf1)/(deg+1)) @ W2 + b2             (V_WMMA f32)
//
// Memory-bound problem (~3.5 GB moved -> ~150us at 23.3 TB/s); GEMM is only
// 4.9 GFLOP, so we keep full fp32 precision via V_WMMA_F32_16X16X4_F32.
// ---------------------------------------------------------------------------

#define FEAT_DIM 128

typedef __attribute__((ext_vector_type(2))) float v2f;
typedef __attribute__((ext_vector_type(8))) float v8f;

// ---------------------------------------------------------------------------
// Edge scatter: one wave32 per edge, each lane moves float4 (32*4 = 128).
// ---------------------------------------------------------------------------
template <bool COUNT_DEG>
__global__ __launch_bounds__(256) void edge_scatter_kernel(
    const float* __restrict__ h,     // [N, 128] source features
    const int* __restrict__ src,     // [E]
    const int* __restrict__ dst,     // [E]
    float* __restrict__ neigh,       // [N, 128] accumulator (pre-zeroed)
    float* __restrict__ deg,         // [N] degree accumulator (pre-zeroed)
    int n_edges)
{
    const int e = blockIdx.x * (blockDim.x >> 5) + (threadIdx.x >> 5);
    if (e >= n_edges) return;
    const int lane = threadIdx.x & 31;

    const int s = src[e];
    const int d = dst[e];

    const float4 v =
        reinterpret_cast<const float4*>(h + (size_t)s * FEAT_DIM)[lane];
    float* base = neigh + (size_t)d * FEAT_DIM + lane * 4;
    unsafeAtomicAdd(base + 0, v.x);
    unsafeAtomicAdd(base + 1, v.y);
    unsafeAtomicAdd(base + 2, v.z);
    unsafeAtomicAdd(base + 3, v.w);
    if (COUNT_DEG && lane == 0) {
        unsafeAtomicAdd(deg + d, 1.0f);
    }
}

// ---------------------------------------------------------------------------
// Fused normalize + GEMM (+bias, optional ReLU) using V_WMMA_F32_16X16X4_F32.
//
// One block handles a 16-node M-tile; OUT_DIM/16 waves each compute one 16x16
// output tile. A-tile (16 x IN_DIM, normalized) staged in LDS (padded stride).
//
// ISA 7.12.2 layouts (wave32, 32-bit):
//   A 16x4 : lane<16 -> row=lane, VGPR0=K0, VGPR1=K1 ; lane>=16 -> K2,K3
//   B 4x16 : lane<16 -> col=lane, VGPR0=K0, VGPR1=K1 ; lane>=16 -> K2,K3
//   C 16x16: VGPR r, lane<16 -> M=r,  N=lane ; lane>=16 -> M=r+8, N=lane-16
//
// BOUNDS=false is dispatched when n_nodes % 16 == 0 (true for N=100000):
// no clamps, unconditional coalesced stores, branch-free epilogue.
// ---------------------------------------------------------------------------
template <int IN_DIM, int OUT_DIM, bool RELU, bool BOUNDS>
__global__ __launch_bounds__((OUT_DIM / 16) * 32) void sage_gemm_kernel(
    const float* __restrict__ neigh,  // [N, IN_DIM] aggregated sums
    const float* __restrict__ self,   // [N, IN_DIM] own features
    const float* __restrict__ deg,    // [N]
    const float* __restrict__ W,      // [IN_DIM, OUT_DIM] row-major
    const float* __restrict__ bias,   // [OUT_DIM]
    float* __restrict__ out,          // [N, OUT_DIM]
    int n_nodes)
{
    constexpr int NWAVES = OUT_DIM / 16;
    constexpr int APAD = IN_DIM + 4;  // LDS row stride 132 words -> no conflicts
    __shared__ float As[16 * APAD];

    const int m0 = blockIdx.x * 16;
    const int tid = threadIdx.x;
    const int lane = tid & 31;
    const int wave = tid >> 5;

    // Cooperative load + normalize of the 16 x IN_DIM A tile.
    for (int i = tid; i < 16 * IN_DIM; i += NWAVES * 32) {
        const int r = i / IN_DIM;
        const int c = i - r * IN_DIM;
        int node = m0 + r;
        if (BOUNDS && node >= n_nodes) node = n_nodes - 1;
        const float scale = 1.0f / (deg[node] + 1.0f);
        As[r * APAD + c] =
            (neigh[(size_t)node * IN_DIM + c] + self[(size_t)node * IN_DIM + c]) * scale;
    }
    __syncthreads();

    const int n0 = wave * 16;
    const int half = lane >> 4;   // 0: K pair {0,1}; 1: K pair {2,3}
    const int l15 = lane & 15;    // A row / B-C column within tile

    v8f acc = {};
    const float* Arow = As + l15 * APAD;
    const float* Wcol = W + n0 + l15;
#pragma unroll 4
    for (int k = 0; k < IN_DIM; k += 4) {
        const int ka = k + 2 * half;
        v2f a, b;
        a.x = Arow[ka];
        a.y = Arow[ka + 1];
        b.x = Wcol[(size_t)ka * OUT_DIM];
        b.y = Wcol[(size_t)(ka + 1) * OUT_DIM];
        acc = __builtin_amdgcn_wmma_f32_16x16x4_f32(
            /*neg_a=*/false, a, /*neg_b=*/false, b,
            /*c_mod=*/(short)0, acc, /*reuse_a=*/false, /*reuse_b=*/false);
    }

    const float bv = bias[n0 + l15];
    float* outp = out + ((size_t)(m0 + half * 8)) * OUT_DIM + n0 + l15;
#pragma unroll
    for (int r = 0; r < 8; ++r) {
        float v = acc[r] + bv;
        if (RELU) v = fmaxf(v, 0.0f);
        if (BOUNDS) {
            const int row = m0 + r + half * 8;
            if (row < n_nodes) out[(size_t)row * OUT_DIM + n0 + l15] = v;
        } else {
            outp[(size_t)r * OUT_DIM] = v;
        }
    }
}

// ---------------------------------------------------------------------------
// Host-side orchestration (graph-capture safe: only kernels + hipMemsetAsync).
// Inputs (setup_inputs order): 0 feat, 1 W1, 2 b1, 3 W2, 4 b2, 5 src, 6 dst.
// ---------------------------------------------------------------------------
extern "C" void kernel_launch(void* const* d_in, const int* in_sizes, int n_in,
                              void* d_out, int out_size, void* d_ws, size_t ws_size,
                              hipStream_t stream)
{
    const float* feat = (const float*)d_in[0];
    const float* W1   = (const float*)d_in[1];
    const float* b1   = (const float*)d_in[2];
    const float* W2   = (const float*)d_in[3];
    const float* b2   = (const float*)d_in[4];
    const int*   src  = (const int*)d_in[5];
    const int*   dst  = (const int*)d_in[6];

    const int n_nodes = in_sizes[0] / FEAT_DIM;   // 100000
    const int n_edges = in_sizes[5];              // 1600000
    const bool exact  = (n_nodes % 16) == 0;      // true for N=100000

    // Workspace layout
    float* deg  = (float*)d_ws;                                  // [N]
    float* buf0 = deg + n_nodes;                                 // [N,128] accum
    float* buf1 = buf0 + (size_t)n_nodes * FEAT_DIM;             // [N,128] h1

    const size_t featBytes = (size_t)n_nodes * FEAT_DIM * sizeof(float);
    const int gemm_grid = (n_nodes + 15) / 16;                   // 6250
    const int scat_grid = (n_edges + 7) / 8;                     // 8 waves/block

    // ---- Layer 1 ----
    hipMemsetAsync(deg, 0, (size_t)n_nodes * sizeof(float), stream);
    hipMemsetAsync(buf0, 0, featBytes, stream);

    edge_scatter_kernel<true><<<scat_grid, 256, 0, stream>>>(
        feat, src, dst, buf0, deg, n_edges);

    if (exact) {
        sage_gemm_kernel<128, 128, true, false><<<gemm_grid, 256, 0, stream>>>(
            buf0, feat, deg, W1, b1, buf1, n_nodes);
    } else {
        sage_gemm_kernel<128, 128, true, true><<<gemm_grid, 256, 0, stream>>>(
            buf0, feat, deg, W1, b1, buf1, n_nodes);
    }

    // ---- Layer 2 ----
    hipMemsetAsync(buf0, 0, featBytes, stream);

    edge_scatter_kernel<false><<<scat_grid, 256, 0, stream>>>(
        buf1, src, dst, buf0, deg, n_edges);

    if (exact) {
        sage_gemm_kernel<128, 64, false, false><<<gemm_grid, 128, 0, stream>>>(
            buf0, buf1, deg, W2, b2, (float*)d_out, n_nodes);
    } else {
        sage_gemm_kernel<128, 64, false, true><<<gemm_grid, 128, 0, stream>>>(
            buf0, buf1, deg, W2, b2, (float*)d_out, n_nodes);
    }
}